// LovaszSoftmaxLoss_82454782148839
// MI455X (gfx1250) — compile-verified
//
#include <hip/hip_runtime.h>
#include <hip/hip_bf16.h>
#include <math.h>

// ---------------------------------------------------------------------------
// Lovasz-Softmax loss, MI455X (gfx1250).
// Sort-free formulation: the Lovasz sum is invariant to ordering within
// equal-error groups, so an exact bucket histogram (count + fg-count per
// 1/65536-wide error bucket) + descending prefix scan reproduces the loss;
// the bucket midpoint stands in for the group error (|err| <= 7.6e-6).
// Kernel A: streaming softmax + ONE packed-u64 atomic per pixel-class
//           (histograms are 10 MB -> fully L2-resident on the 192 MB L2).
// Kernel B: per-class 65536-bucket scan; the 256-chunk-total prefix scan runs
//           on tensor cores via triangular matmuls (V_WMMA_F32_16X16X4_F32).
// Kernel C: masked mean over present classes.
// ---------------------------------------------------------------------------

#define IGNORE_INDEX (-100)
#define NCLS 19
#define BATCH 8
#define HDIM 512
#define WDIM 512
#define HW (HDIM * WDIM)
#define NPIX (BATCH * HW)
#define NB 65536           // error buckets per class (key = floor(e * 65536))
#define CHUNK 256          // buckets per thread in kernel B
#define BLKB 256           // threads per class-block in kernel B

typedef float v8f __attribute__((ext_vector_type(8)));
typedef float v2f __attribute__((ext_vector_type(2)));

// ---------------------------------------------------------------------------
// Kernel A: per-pixel softmax over 19 classes + histogram scatter.
// logits layout [B, C, H, W]: consecutive threads walk consecutive hw, so each
// class-plane load is a fully coalesced 128B/wave burst.
// ---------------------------------------------------------------------------
__global__ __launch_bounds__(256) void k_softmax_hist(
    const float* __restrict__ logits, const int* __restrict__ labels,
    unsigned long long* __restrict__ packed)  // [C][NB] {fg:hi32, cnt:lo32}
{
  const int n = blockIdx.x * 256 + threadIdx.x;
  if (n >= NPIX) return;
  const int b  = n / HW;
  const int hw = n - b * HW;
  const float* base = logits + ((size_t)b * NCLS) * HW + hw;
  __builtin_prefetch(base, 0, 1);  // -> global_prefetch_b8

  float l[NCLS];
  float mx = -3.4e38f;
#pragma unroll
  for (int c = 0; c < NCLS; ++c) {
    l[c] = base[(size_t)c * HW];
    mx = fmaxf(mx, l[c]);
  }
  float s = 0.f;
#pragma unroll
  for (int c = 0; c < NCLS; ++c) {
    l[c] = __expf(l[c] - mx);
    s += l[c];
  }
  const float inv = __frcp_rn(s);

  const int lb = labels[n];
  const int valid = (lb != IGNORE_INDEX);

#pragma unroll
  for (int c = 0; c < NCLS; ++c) {
    const float p = l[c] * inv;
    const int isfg = valid && (lb == c);
    const float e = valid ? fabsf((isfg ? 1.f : 0.f) - p) : 0.f;
    int bk = (int)(e * 65536.0f);
    bk = (bk > NB - 1) ? (NB - 1) : bk;
    // one 64-bit atomic carries both the count (+1, lo) and fg count (hi)
    atomicAdd(&packed[(size_t)c * NB + (size_t)bk],
              1ull | ((unsigned long long)isfg << 32));
  }
}

// ---------------------------------------------------------------------------
// WMMA helper: D = A(16x4) * B(4x16) + C(16x16), f32.
// ---------------------------------------------------------------------------
__device__ __forceinline__ v8f wmma_f32_16x16x4(v2f a, v2f b, v8f c) {
  return __builtin_amdgcn_wmma_f32_16x16x4_f32(
      /*neg_a=*/false, a, /*neg_b=*/false, b,
      /*c_mod=*/(short)0, c, /*reuse_a=*/false, /*reuse_b=*/false);
}

// ---------------------------------------------------------------------------
// Inclusive row-major cumsum of 256 f32 values via tensor cores:
//   incl = X*U + (Lstrict*X)*Ones        (all 16x16)
// with U = upper-tri ones (incl diag), Lstrict = strictly-lower-tri ones.
// K=16 is split into 4 chunks of 4 for V_WMMA_F32_16X16X4_F32.
// ISA operand maps (wave32):
//   A 16x4 : lanes 0-15 M=lane, {v0,v1}={K0,K1}; lanes 16-31 {K2,K3}
//   B 4x16 : lanes 0-15 N=lane, {v0,v1}={K0,K1}; lanes 16-31 {K2,K3}
//   C 16x16: lanes 0-15 M=reg,  N=lane; lanes 16-31 M=reg+8, N=lane-16
// Executed by wave 0; all threads of the block must call (internal barriers).
// Counts < 2^24 so f32 accumulation is exact.
// ---------------------------------------------------------------------------
__device__ __forceinline__ void scan256_wmma(const float* __restrict__ src,
                                             float* __restrict__ dst,
                                             float* __restrict__ scratch)
{
  const int t = threadIdx.x;
  const int lane = t & 31;
  const int half = lane >> 4;   // 0: lanes 0-15, 1: lanes 16-31
  const int m = lane & 15;      // A-row index / B-col index / C-col index

  v8f acc = {0.f, 0.f, 0.f, 0.f, 0.f, 0.f, 0.f, 0.f};

  if (t < 32) {                 // whole wave active -> EXEC all ones
    v8f r2 = {0.f, 0.f, 0.f, 0.f, 0.f, 0.f, 0.f, 0.f};
#pragma unroll
    for (int kc = 0; kc < 4; ++kc) {
      const int k0 = kc * 4 + half * 2;  // first K this lane-half supplies
      // acc += X[:, k0..] * U[k0.., :]   (row-wise inclusive prefix)
      v2f a, bu;
      a.x = src[m * 16 + k0];
      a.y = src[m * 16 + k0 + 1];
      bu.x = (k0     <= m) ? 1.f : 0.f;  // U[k0][n],   n == m
      bu.y = (k0 + 1 <= m) ? 1.f : 0.f;  // U[k0+1][n]
      acc = wmma_f32_16x16x4(a, bu, acc);
      // r2 += Lstrict[:, k0..] * X[k0.., :]  (sum of previous rows, per col)
      v2f a2, b2;
      a2.x = (k0     < m) ? 1.f : 0.f;   // Lstrict[m][k0]
      a2.y = (k0 + 1 < m) ? 1.f : 0.f;
      b2.x = src[(k0)     * 16 + m];     // X[k0][n]
      b2.y = src[(k0 + 1) * 16 + m];
      r2 = wmma_f32_16x16x4(a2, b2, r2);
    }
    // spill r2 (C layout) row-major so it can be re-read in A layout
#pragma unroll
    for (int r = 0; r < 8; ++r) scratch[(r + half * 8) * 16 + m] = r2[r];
  }
  __syncthreads();
  if (t < 32) {
#pragma unroll
    for (int kc = 0; kc < 4; ++kc) {
      const int k0 = kc * 4 + half * 2;
      // acc += R2[:, k0..] * Ones[k0.., :]  (broadcast row-block totals)
      v2f a, b1;
      a.x = scratch[m * 16 + k0];
      a.y = scratch[m * 16 + k0 + 1];
      b1.x = 1.f;
      b1.y = 1.f;
      acc = wmma_f32_16x16x4(a, b1, acc);
    }
#pragma unroll
    for (int r = 0; r < 8; ++r) dst[(r + half * 8) * 16 + m] = acc[r];
  }
  __syncthreads();
}

// ---------------------------------------------------------------------------
// Kernel B: one block per class. Buckets walked in DESCENDING error order.
// Phase 1: 256 threads x 256-bucket chunk sums -> (K,F) chunk totals.
// Phase 2: WMMA inclusive scan of chunk totals.
// Phase 3: per-bucket Jaccard telescoping:  sum_g emid_g * (J(K_g)-J(K_{g-1}))
//          with J(K,F) = 1 - (G-F)/(G+K-F), exact under tie-reordering.
// ---------------------------------------------------------------------------
__global__ __launch_bounds__(BLKB) void k_lovasz_class(
    const unsigned long long* __restrict__ packed,
    float* __restrict__ loss_c, float* __restrict__ present_c)
{
  __shared__ float sK[BLKB], sF[BLKB], sKi[BLKB], sFi[BLKB];
  __shared__ float scr[BLKB], red[BLKB];

  const int c = blockIdx.x;
  const int t = threadIdx.x;
  const unsigned long long* pk = packed + (size_t)c * NB;

  const int b0 = NB - 1 - t * CHUNK;  // this chunk's first (largest-e) bucket

  float ck = 0.f, cf = 0.f;
  for (int j = 0; j < CHUNK; ++j) {
    const unsigned long long p = pk[b0 - j];
    ck += (float)(unsigned)(p & 0xffffffffu);
    cf += (float)(unsigned)(p >> 32);
  }
  sK[t] = ck;
  sF[t] = cf;
  __syncthreads();

  scan256_wmma(sK, sKi, scr);
  scan256_wmma(sF, sFi, scr);

  const float G = sFi[BLKB - 1];  // total foreground count for this class
  float lsum = 0.f;
  if (G > 0.f) {
    float runK = sKi[t] - ck;     // exclusive prefixes for this chunk
    float runF = sFi[t] - cf;
    float Jprev = 1.f - (G - runF) / (G + runK - runF);  // K=0 -> J=0
    for (int j = 0; j < CHUNK; ++j) {
      const int bkt = b0 - j;
      const unsigned long long p = pk[bkt];
      const float cn = (float)(unsigned)(p & 0xffffffffu);
      if (cn > 0.f) {
        runK += cn;
        runF += (float)(unsigned)(p >> 32);
        const float J = 1.f - (G - runF) / (G + runK - runF);
        const float emid = ((float)bkt + 0.5f) * (1.0f / 65536.0f);
        lsum += emid * (J - Jprev);  // group error * telescoped Jaccard delta
        Jprev = J;
      }
    }
  }
  red[t] = lsum;
  __syncthreads();
  for (int off = BLKB / 2; off > 0; off >>= 1) {
    if (t < off) red[t] += red[t + off];
    __syncthreads();
  }
  if (t == 0) {
    loss_c[c] = red[0];
    present_c[c] = (G > 0.f) ? 1.f : 0.f;
  }
}

// ---------------------------------------------------------------------------
// Kernel C: masked mean over present classes (single wave32).
// ---------------------------------------------------------------------------
__global__ void k_finalize(const float* __restrict__ loss_c,
                           const float* __restrict__ present_c,
                           float* __restrict__ out)
{
  const int t = threadIdx.x;
  float s = 0.f, np = 0.f;
  if (t < NCLS) {
    s = loss_c[t] * present_c[t];
    np = present_c[t];
  }
  for (int off = 16; off > 0; off >>= 1) {
    s += __shfl_down(s, off, 32);
    np += __shfl_down(np, off, 32);
  }
  if (t == 0) out[0] = s / fmaxf(np, 1.f);
}

// ---------------------------------------------------------------------------
extern "C" void kernel_launch(void* const* d_in, const int* in_sizes, int n_in,
                              void* d_out, int out_size, void* d_ws, size_t ws_size,
                              hipStream_t stream)
{
  (void)in_sizes; (void)n_in; (void)out_size; (void)ws_size;
  const float* logits = (const float*)d_in[0];
  const int* labels = (const int*)d_in[1];
  float* out = (float*)d_out;

  char* ws = (char*)d_ws;
  const size_t packed_bytes = (size_t)NCLS * NB * sizeof(unsigned long long);
  unsigned long long* packed = (unsigned long long*)ws;
  float* loss_c = (float*)(ws + packed_bytes);
  float* present_c = loss_c + NCLS;

  // histograms must be zero on every call (graph replays reuse d_ws)
  hipMemsetAsync(d_ws, 0, packed_bytes, stream);

  k_softmax_hist<<<NPIX / 256, 256, 0, stream>>>(logits, labels, packed);
  k_lovasz_class<<<NCLS, BLKB, 0, stream>>>(packed, loss_c, present_c);
  k_finalize<<<1, 32, 0, stream>>>(loss_c, present_c, out);
}